// VGGT_38156489458369
// MI455X (gfx1250) — compile-verified
//
#include <hip/hip_runtime.h>
#include <hip/hip_bf16.h>

typedef __attribute__((ext_vector_type(2))) float v2f;
typedef __attribute__((ext_vector_type(8))) float v8f;

#define EMBED 256
#define PDIM  768
#define KCH   96                    // K-chunk per LDS buffer
#define NCHUNK (PDIM / KCH)         // 8
#define STR   100                   // padded LDS row stride (floats); conflict-free
#define SLABF (16 * STR)            // floats per buffer
#define SLABBYTES (SLABF * 4)
#define L4    ((16 * KCH / 4) / 32) // float4 async loads per lane per chunk = 12
#define WAVES_PER_BLOCK 4
#define TOKENS_PER_BLOCK (WAVES_PER_BLOCK * 16)

// ---------------------------------------------------------------------------
// Kernel 1: fold the three linear layers into one fused projection.
//   v_tok[k] = sum_e W_tok[k,e] * W_depth[e]      k in [0,768)
//   v_pos[j] = sum_e W_pos[j,e] * W_depth[e]      j in [0,3)
//   c        = sum_e (b_tok[e]+b_pos[e]) * W_depth[e] + b_depth
// ---------------------------------------------------------------------------
__global__ __launch_bounds__(256) void vggt_fold_kernel(
    const float* __restrict__ Wtok, const float* __restrict__ btok,
    const float* __restrict__ Wpos, const float* __restrict__ bpos,
    const float* __restrict__ Wdep, const float* __restrict__ bdep,
    float* __restrict__ ws)
{
  int k = blockIdx.x * blockDim.x + threadIdx.x;
  if (k < PDIM) {
    const float* row = Wtok + (size_t)k * EMBED;
    float s = 0.f;
    #pragma unroll 4
    for (int e = 0; e < EMBED; ++e) s += row[e] * Wdep[e];
    ws[k] = s;
  } else if (k < PDIM + 3) {
    int j = k - PDIM;
    const float* row = Wpos + (size_t)j * EMBED;
    float s = 0.f;
    for (int e = 0; e < EMBED; ++e) s += row[e] * Wdep[e];
    ws[PDIM + j] = s;
  } else if (k == PDIM + 3) {
    float s = bdep[0];
    for (int e = 0; e < EMBED; ++e) s += (btok[e] + bpos[e]) * Wdep[e];
    ws[PDIM + 3] = s;
  }
}

// ---------------------------------------------------------------------------
// Kernel 2: memory-bound fused GEMV via V_WMMA_F32_16X16X4_F32 with
// double-buffered async HBM->LDS staging (GLOBAL_LOAD_ASYNC_TO_LDS_B128,
// ASYNCcnt-tracked). One wave owns a 16-token tile and two private LDS
// buffers -> no block barriers in the main loop.
// ---------------------------------------------------------------------------
__global__ __launch_bounds__(WAVES_PER_BLOCK * 32) void vggt_depth_kernel(
    const float* __restrict__ patches, const float* __restrict__ coords,
    const float* __restrict__ ws, float* __restrict__ out, int T)
{
  __shared__ __align__(16) float ldsA[WAVES_PER_BLOCK * 2 * SLABF];
  __shared__ __align__(16) float vsh[PDIM + 4];

  const int tid  = threadIdx.x;
  const int wave = tid >> 5;        // wave32 on gfx1250
  const int lane = tid & 31;
  const int hi   = lane >> 4;       // 0 -> K pair {0,1}, 1 -> K pair {2,3}
  const int m    = lane & 15;       // tile row (M) for A-fragment reads

  // Stage fused vectors (v_tok | v_pos | c) once per block.
  for (int i = tid; i < PDIM + 4; i += WAVES_PER_BLOCK * 32) vsh[i] = ws[i];

  const int tileBase = (blockIdx.x * WAVES_PER_BLOCK + wave) * 16;
  float* const slab0 = &ldsA[wave * 2 * SLABF];
  const unsigned slab0B = (unsigned)(size_t)slab0;   // AS(3) offset in low bits

  // Per-lane staging addresses (fixed for all chunks; +kc / +buf applied later).
  const float* gb[L4];
  unsigned     lb[L4];
  #pragma unroll
  for (int i = 0; i < L4; ++i) {
    int f   = i * 32 + lane;            // float4 index within a chunk
    int row = f / (KCH / 4);
    int c4  = f % (KCH / 4);
    int rg  = tileBase + row;
    if (rg >= T) rg = T - 1;            // tail tiles: clamp (stores are guarded)
    gb[i] = patches + (size_t)rg * PDIM + c4 * 4;
    lb[i] = slab0B + (unsigned)(row * STR + c4 * 4) * 4u;
  }

  // Prefetch chunk 0 into buffer 0 (direct HBM -> LDS, no VGPR round trip).
  #pragma unroll
  for (int i = 0; i < L4; ++i) {
    asm volatile("global_load_async_to_lds_b128 %0, %1, off"
                 :: "v"(lb[i]), "v"(gb[i]) : "memory");
  }

  __syncthreads();                      // vsh visible to all waves

  v8f acc = {0.f, 0.f, 0.f, 0.f, 0.f, 0.f, 0.f, 0.f};

  #pragma unroll
  for (int c = 0; c < NCHUNK; ++c) {
    const int buf = c & 1;

    if (c + 1 < NCHUNK) {
      // Issue next chunk's loads before consuming this one (overlap).
      const int nbuf = (c + 1) & 1;
      #pragma unroll
      for (int i = 0; i < L4; ++i) {
        const float* g = gb[i] + (c + 1) * KCH;
        unsigned     l = lb[i] + (unsigned)(nbuf * SLABBYTES);
        asm volatile("global_load_async_to_lds_b128 %0, %1, off"
                     :: "v"(l), "v"(g) : "memory");
      }
      // Async loads complete in order: <= L4 outstanding => chunk c landed.
      asm volatile("s_wait_asynccnt %0" :: "i"(L4) : "memory");
    } else {
      asm volatile("s_wait_asynccnt 0x0" ::: "memory");
    }

    // 24 chained f32 WMMAs on this chunk (192 over full K=768).
    float* const myA = slab0 + buf * SLABF;
    const int kc = c * KCH;
    #pragma unroll
    for (int kk = 0; kk < KCH; kk += 4) {
      v2f a, b;
      const float* ap = myA + m * STR + kk + 2 * hi;   // conflict-free ds_load_b64
      a[0] = ap[0];
      a[1] = ap[1];
      const float* bp = vsh + kc + kk + 2 * hi;        // broadcast read
      b[0] = bp[0];
      b[1] = bp[1];
      acc = __builtin_amdgcn_wmma_f32_16x16x4_f32(
          /*neg_a=*/false, a, /*neg_b=*/false, b,
          /*c_mod=*/(short)0, acc, /*reuse_a=*/false, /*reuse_b=*/false);
    }

    // Retire this buffer's LDS reads before the async refill two chunks out
    // can overwrite it (WAR between DS reads and async LDS writes).
    asm volatile("s_wait_dscnt 0x0" ::: "memory");
  }

  // D layout: VGPR j holds row M=j (lanes 0-15) / M=j+8 (lanes 16-31), all
  // columns identical. Lane 0 stores tokens base+0..7, lane 16 base+8..15.
  if (m == 0) {
    const float v0 = vsh[PDIM], v1 = vsh[PDIM + 1];
    const float v2 = vsh[PDIM + 2], cc = vsh[PDIM + 3];
    const int rowBase = tileBase + hi * 8;
    #pragma unroll
    for (int j = 0; j < 8; ++j) {
      int t = rowBase + j;
      if (t < T) {
        const float* cp = coords + (size_t)t * 3;
        out[t] = acc[j] + cp[0] * v0 + cp[1] * v1 + cp[2] * v2 + cc;
      }
    }
  }
}

// ---------------------------------------------------------------------------
extern "C" void kernel_launch(void* const* d_in, const int* in_sizes, int n_in,
                              void* d_out, int out_size, void* d_ws, size_t ws_size,
                              hipStream_t stream) {
  // setup_inputs order:
  // 0 counts(i64, unused: scatter+gather is identity), 1 all_coords[T,3],
  // 2 all_patches[T,768], 3 W_tok[768,256], 4 b_tok[256], 5 W_pos[3,256],
  // 6 b_pos[256], 7 W_depth[256,1], 8 b_depth[1]
  const float* coords  = (const float*)d_in[1];
  const float* patches = (const float*)d_in[2];
  const float* Wtok    = (const float*)d_in[3];
  const float* btok    = (const float*)d_in[4];
  const float* Wpos    = (const float*)d_in[5];
  const float* bpos    = (const float*)d_in[6];
  const float* Wdep    = (const float*)d_in[7];
  const float* bdep    = (const float*)d_in[8];
  float* out = (float*)d_out;
  float* ws  = (float*)d_ws;   // 772 floats of fused-projection scratch

  const int T = in_sizes[2] / PDIM;   // 39904 for the reference counts

  vggt_fold_kernel<<<(PDIM + 4 + 255) / 256, 256, 0, stream>>>(
      Wtok, btok, Wpos, bpos, Wdep, bdep, ws);

  const int blocks = (T + TOKENS_PER_BLOCK - 1) / TOKENS_PER_BLOCK;
  vggt_depth_kernel<<<blocks, WAVES_PER_BLOCK * 32, 0, stream>>>(
      patches, coords, ws, out, T);
}